// SelfAttentionOpCUDA_79972291052155
// MI455X (gfx1250) — compile-verified
//
#include <hip/hip_runtime.h>
#include <math.h>

#define D_DIM 512
#define RANK  128
#define BATCH 512
#define NB    2        // batch elements per block in the fused kernel

typedef __attribute__((ext_vector_type(2))) float v2f;
typedef __attribute__((ext_vector_type(8))) float v8f;

// ---------------------------------------------------------------------------
// Kernel 1: G = (Q_emb @ K_emb^T) * (1/sqrt(RANK)), [512 x 512] fp32.
// V_WMMA_F32_16X16X4_F32 (CDNA5 fp32 matrix path, exact precision).
// 1024 16x16 tiles; 4 waves per block, one tile per wave, 32 WMMAs per tile.
// ---------------------------------------------------------------------------
__global__ void __launch_bounds__(128)
gemm_G_wmma(const float* __restrict__ Qe, const float* __restrict__ Ke,
            float* __restrict__ G) {
    const int wave = threadIdx.x >> 5;
    const int lane = threadIdx.x & 31;
    const int tile = blockIdx.x * 4 + wave;      // 0..1023
    const int i0 = (tile >> 5) * 16;             // row of G tile
    const int j0 = (tile & 31) * 16;             // col of G tile

    const int hlf = lane >> 4;                   // 0: K={k,k+1}, 1: K={k+2,k+3}
    const int m   = lane & 15;                   // row within tile

    // A: 16x4 tile of Q_emb rows i0..i0+15.  B: 4x16 tile = K_emb rows j0..j0+15
    // transposed; both use identical per-lane addressing (contiguous float2).
    const float* arow = Qe + (i0 + m) * RANK + 2 * hlf;
    const float* brow = Ke + (j0 + m) * RANK + 2 * hlf;

    v8f c = {};
#pragma unroll 8
    for (int k = 0; k < RANK; k += 4) {
        v2f a = *(const v2f*)(arow + k);
        v2f b = *(const v2f*)(brow + k);
        c = __builtin_amdgcn_wmma_f32_16x16x4_f32(
                /*neg_a=*/false, a, /*neg_b=*/false, b,
                /*c_mod=*/(short)0, c, /*reuse_a=*/false, /*reuse_b=*/false);
    }

    const float inv_sqrt_rank = 0.08838834764831845f;  // 1/sqrt(128)
#pragma unroll
    for (int p = 0; p < 8; ++p) {
        int row = i0 + p + 8 * hlf;
        int col = j0 + m;
        G[row * D_DIM + col] = c[p] * inv_sqrt_rank;
    }
}

// ---------------------------------------------------------------------------
// Kernel 2: w[j] = V_emb[j,:] . out_proj   (512 independent length-128 dots)
// ---------------------------------------------------------------------------
__global__ void __launch_bounds__(256)
compute_w(const float* __restrict__ Ve, const float* __restrict__ op,
          float* __restrict__ w) {
    int j = blockIdx.x * blockDim.x + threadIdx.x;
    if (j >= D_DIM) return;
    const float4* v4 = (const float4*)(Ve + j * RANK);
    const float4* o4 = (const float4*)op;
    float s = 0.f;
#pragma unroll 8
    for (int r = 0; r < RANK / 4; ++r) {
        float4 a = v4[r], b = o4[r];
        s += a.x * b.x + a.y * b.y + a.z * b.z + a.w * b.w;
    }
    w[j] = s;
}

// ---------------------------------------------------------------------------
// Kernel 3: fused score + exact two-pass softmax + weighted reduce.
//   score[i,j] = x_i * x_j * G[i,j];  out_i = x_i + gate * softmax_j . (x_j w_j)
// Each block owns NB batch elements (amortizes the G stream over NB batches).
// 8 waves; wave owns rows i = wave, wave+8, ...; lanes split j (coalesced).
// G row, x_j, w_j and scores are register-resident (16 VGPRs each, wave32);
// two-pass softmax costs exactly ONE v_exp_f32 per score element.
// ---------------------------------------------------------------------------
__global__ void __launch_bounds__(256)
attn_fused(const float* __restrict__ x, const float* __restrict__ G,
           const float* __restrict__ w, const float* __restrict__ gate_logit,
           float* __restrict__ out) {
    __shared__ float xs[NB * D_DIM];
    __shared__ float ws[D_DIM];

    const int b0   = blockIdx.x * NB;
    const int tid  = threadIdx.x;
    const int lane = tid & 31;
    const int wave = tid >> 5;

    for (int j = tid; j < D_DIM; j += 256) {
        ws[j] = w[j];
#pragma unroll
        for (int n = 0; n < NB; ++n)
            xs[n * D_DIM + j] = x[(b0 + n) * D_DIM + j];
    }
    __syncthreads();

    const float gate = 1.0f / (1.0f + __expf(-gate_logit[0]));

    // per-lane register cache of w[j] for j = lane, lane+32, ...
    float wreg[16];
#pragma unroll
    for (int u = 0; u < 16; ++u) wreg[u] = ws[lane + 32 * u];

    for (int i = wave; i < D_DIM; i += 8) {
        // register cache of G row i (coalesced: lanes read consecutive floats)
        const float* Gi = G + i * D_DIM;
        float g[16];
#pragma unroll
        for (int u = 0; u < 16; ++u) g[u] = Gi[lane + 32 * u];

#pragma unroll
        for (int n = 0; n < NB; ++n) {
            const float* xb = xs + n * D_DIM;
            const float xi  = xb[i];

            // pass 1: scores into registers + exact row max
            float sc[16], xj[16];
            float mx = -INFINITY;
#pragma unroll
            for (int u = 0; u < 16; ++u) {
                float xv = xb[lane + 32 * u];
                float t  = xi * xv * g[u];
                xj[u] = xv;
                sc[u] = t;
                mx = fmaxf(mx, t);
            }
#pragma unroll
            for (int off = 16; off >= 1; off >>= 1)
                mx = fmaxf(mx, __shfl_xor(mx, off, 32));

            // pass 2: one exp per element, accumulate sum and weighted sum
            float s = 0.f, a = 0.f;
#pragma unroll
            for (int u = 0; u < 16; ++u) {
                float e = __expf(sc[u] - mx);
                s += e;
                a += e * (xj[u] * wreg[u]);
            }
#pragma unroll
            for (int off = 16; off >= 1; off >>= 1) {
                s += __shfl_xor(s, off, 32);
                a += __shfl_xor(a, off, 32);
            }

            if (lane == 0)
                out[(b0 + n) * D_DIM + i] = xi + gate * (a / s);
        }
    }
}

// ---------------------------------------------------------------------------
extern "C" void kernel_launch(void* const* d_in, const int* in_sizes, int n_in,
                              void* d_out, int out_size, void* d_ws, size_t ws_size,
                              hipStream_t stream) {
    const float* x   = (const float*)d_in[0];  // [B, D]
    const float* Qe  = (const float*)d_in[1];  // [D, RANK]
    const float* Ke  = (const float*)d_in[2];  // [D, RANK]
    const float* Ve  = (const float*)d_in[3];  // [D, RANK]
    const float* op  = (const float*)d_in[4];  // [RANK]
    const float* gl  = (const float*)d_in[5];  // scalar
    float* out = (float*)d_out;                // [B, D]

    float* G = (float*)d_ws;                               // 512*512*4 = 1 MB
    float* w = (float*)((char*)d_ws + (size_t)D_DIM * D_DIM * sizeof(float)); // 2 KB

    // 1) G = Qe @ Ke^T / sqrt(rank)   (WMMA fp32)
    gemm_G_wmma<<<256, 128, 0, stream>>>(Qe, Ke, G);
    // 2) w = Ve @ out_proj
    compute_w<<<2, 256, 0, stream>>>(Ve, op, w);
    // 3) fused per-batch softmax-attention, NB batches per block
    attn_fused<<<BATCH / NB, 256, 0, stream>>>(x, G, w, gl, out);
}